// GaussianRegularizationLoss_84971632984448
// MI455X (gfx1250) — compile-verified
//
#include <hip/hip_runtime.h>
#include <hip/hip_bf16.h>
#include <math.h>

// ---------------------------------------------------------------------------
// GaussianRegularizationLoss for MI455X (gfx1250, wave32).
//
// quad(y,x) = a*dx^2 + b*dx*dy + c*dy^2 is rank-3 in (row, col), so each
// 16x16 subtile of the 64x64 field is one V_WMMA_F32_16X16X4_F32:
//   A(m,:) = [c*dy_m^2, 1, dy_m, 0]   (16x4)
//   B(:,n) = [1, a*dx_n^2, b*dx_n, 0] (4x16)
//   D = A x B = quad subtile          (16x16 f32, 8 VGPRs/lane)
// The 2*pi*sqrt(det) factor cancels under per-tile normalization, and
// xlogy(h,h) - h*log(p+eps) == h*(log(h) - log(p+eps)) since h > 0.
//
// Data movement: the 16KB heatmap tile is streamed to LDS with
// GLOBAL_LOAD_ASYNC_TO_LDS_B128 (ASYNCcnt) issued before pass-1 compute,
// so the DMA overlaps WMMA + exp + the S reduction; the async wait is
// folded into the barrier the reduction needs anyway.
// ---------------------------------------------------------------------------

typedef __attribute__((ext_vector_type(2))) float v2f;
typedef __attribute__((ext_vector_type(8))) float v8f;

#define B_DIM   64
#define P_DIM   68
#define HW      64
#define N_TILES (B_DIM * P_DIM)   // 4352 workgroups, one per (b,p) tile
#define EPS     1e-5f

__device__ __forceinline__ float wave_reduce_add(float v) {
  // wave32: 5-step butterfly
  #pragma unroll
  for (int off = 16; off > 0; off >>= 1)
    v += __shfl_xor(v, off, 32);
  return v;
}

__global__ __launch_bounds__(256)
void gauss_kld_tile_kernel(const float* __restrict__ hm,
                           const float* __restrict__ means,
                           const float* __restrict__ cov,
                           float* __restrict__ tile_loss) {
  const int tile = blockIdx.x;
  const int tid  = threadIdx.x;
  const int wave = tid >> 5;
  const int lane = tid & 31;
  const int lm   = lane & 15;     // M (for A) / N (for B) index
  const int half = lane >> 4;     // selects K={0,1} vs K={2,3} pair

  __shared__ float lds_hm[HW * HW];   // 16 KB heatmap tile
  __shared__ float red[8];

  const float* __restrict__ hmT = hm + (size_t)tile * (HW * HW);

  // ---- Kick off the async DMA of the heatmap tile into LDS ----
  // 4 rounds x 256 lanes x 16B = 16KB, fully coalesced b128 streams.
  {
    const float* gsrc = hmT + tid * 4;
    unsigned     ldst = (unsigned)(size_t)(&lds_hm[tid * 4]);
    #pragma unroll
    for (int rnd = 0; rnd < 4; ++rnd) {
      asm volatile("global_load_async_to_lds_b128 %0, %1, off"
                   :: "v"(ldst + (unsigned)(rnd * 4096)),
                      "v"(gsrc + rnd * 1024)
                   : "memory");
    }
  }

  // Uniform per-tile scalars (lowered to SMEM loads).
  const float mx  = means[tile * 2 + 0];
  const float my  = means[tile * 2 + 1];
  const float s00 = cov[tile * 4 + 0];
  const float s01 = cov[tile * 4 + 1];
  const float s10 = cov[tile * 4 + 2];
  const float s11 = cov[tile * 4 + 3];
  const float det = s00 * s11 - s01 * s10 + EPS;
  const float inv_det = 1.0f / det;
  const float ca =  s11 * inv_det;           // dx^2 coefficient
  const float cb = -(s01 + s10) * inv_det;   // dx*dy coefficient
  const float cc =  s00 * inv_det;           // dy^2 coefficient

  float e[2][8];   // exp(-quad/2) per pixel, resident across both passes
  float psum = 0.0f;

  #pragma unroll
  for (int i = 0; i < 2; ++i) {
    const int st     = wave * 2 + i;      // subtile 0..15
    const int rowBlk = (st >> 2) * 16;
    const int colBlk = (st & 3)  * 16;

    // A: 16x4 f32 (2 VGPRs). M = lane&15; lanes 0-15 hold K=0,1;
    // lanes 16-31 hold K=2,3 (per ISA 32-bit A 16x4 layout).
    const float dy = (float)(rowBlk + lm) - my;
    v2f Am;
    Am.x = half ? dy   : cc * dy * dy;    // K=0 | K=2
    Am.y = half ? 0.0f : 1.0f;            // K=1 | K=3

    // B: 4x16 f32 (2 VGPRs), mirrored layout: N = lane&15, K split likewise.
    const float dx = (float)(colBlk + lm) - mx;
    v2f Bm;
    Bm.x = half ? cb * dx : 1.0f;         // K=2 | K=0
    Bm.y = half ? 0.0f    : ca * dx * dx; // K=3 | K=1

    v8f acc = {};
    acc = __builtin_amdgcn_wmma_f32_16x16x4_f32(
        /*neg_a=*/false, Am, /*neg_b=*/false, Bm,
        /*c_mod=*/(short)0, acc, /*reuse_a=*/false, /*reuse_b=*/false);

    #pragma unroll
    for (int r = 0; r < 8; ++r) {
      const float ev = __expf(-0.5f * acc[r]);
      e[i][r] = ev;
      psum += ev;
    }
  }

  // --- Workgroup reduction of S = sum(exp(-quad/2)) over 4096 pixels.
  //     Fold the async-DMA fence into the barrier the reduction needs. ---
  float wsum = wave_reduce_add(psum);
  if (lane == 0) red[wave] = wsum;
  asm volatile("s_wait_asynccnt 0" ::: "memory");   // tile now fully in LDS
  __syncthreads();
  float S = 0.0f;
  #pragma unroll
  for (int wv = 0; wv < 8; ++wv) S += red[wv];      // uniform broadcast read
  const float invS = 1.0f / S;

  // --- Pass 2: KL terms. hm comes from LDS in the WMMA C/D layout.
  //     log-difference form avoids the precise-division macro. ---
  float lsum = 0.0f;
  #pragma unroll
  for (int i = 0; i < 2; ++i) {
    const int st     = wave * 2 + i;
    const int rowBlk = (st >> 2) * 16;
    const int colBlk = (st & 3)  * 16;
    const int row0   = rowBlk + half * 8;
    const int col    = colBlk + lm;
    #pragma unroll
    for (int r = 0; r < 8; ++r) {
      const float hv = lds_hm[(row0 + r) * HW + col];
      const float pr = e[i][r] * invS + EPS;
      lsum += hv * (__logf(hv) - __logf(pr));
    }
  }

  __syncthreads();                 // red[] reuse
  float wl = wave_reduce_add(lsum);
  if (lane == 0) red[wave] = wl;
  __syncthreads();
  if (tid == 0) {
    float t = 0.0f;
    #pragma unroll
    for (int wv = 0; wv < 8; ++wv) t += red[wv];
    tile_loss[tile] = t;
  }
}

// Deterministic final mean over 4352 per-tile losses (single block, fixed
// summation order -> bitwise-stable across graph replays; no float atomics).
__global__ __launch_bounds__(256)
void final_reduce_kernel(const float* __restrict__ tile_loss,
                         float* __restrict__ out) {
  const int tid = threadIdx.x;
  float s = 0.0f;
  for (int t = tid; t < N_TILES; t += 256) s += tile_loss[t];
  __shared__ float red[8];
  float w = wave_reduce_add(s);
  if ((tid & 31) == 0) red[tid >> 5] = w;
  __syncthreads();
  if (tid == 0) {
    float tot = 0.0f;
    #pragma unroll
    for (int wv = 0; wv < 8; ++wv) tot += red[wv];
    out[0] = tot * (1.0f / (float)N_TILES);
  }
}

extern "C" void kernel_launch(void* const* d_in, const int* in_sizes, int n_in,
                              void* d_out, int out_size, void* d_ws, size_t ws_size,
                              hipStream_t stream) {
  const float* hm    = (const float*)d_in[0];  // (B,P,64,64) f32
  const float* means = (const float*)d_in[1];  // (B,P,2)     f32
  const float* cov   = (const float*)d_in[2];  // (B,P,4)     f32
  float* tile_loss   = (float*)d_ws;           // 4352 floats of scratch
  float* out         = (float*)d_out;          // scalar loss

  gauss_kld_tile_kernel<<<N_TILES, 256, 0, stream>>>(hm, means, cov, tile_loss);
  final_reduce_kernel<<<1, 256, 0, stream>>>(tile_loss, out);
}